// Propagator_463856468318
// MI455X (gfx1250) — compile-verified
//
#include <hip/hip_runtime.h>

// ---------------------------------------------------------------------------
// AFQMC-style propagator for MI455X (gfx1250, wave32)
//   WMMA f32 16x16x4 for all propagation GEMMs, TDM (tensor_load_to_lds) to
//   stage the 256x256 operator matrix into LDS once per expm kernel.
//   nbasis=256, nelec=64/spin (packed 128 cols), nfield=512, nts=20
// ---------------------------------------------------------------------------

#define NB 256      // nbasis
#define NE 64       // nelec per spin
#define NC 128      // packed columns (2 spins)
#define NF 512      // nfield
#define NT 20       // time steps
#define EXPM_ORDER 6
#define LDP 65      // padded LDS pitch for ortho panel
#define LDA 257     // padded LDS pitch for staged A (257 mod 64 == 1 -> no conflicts)

typedef __attribute__((ext_vector_type(2))) float v2f;
typedef __attribute__((ext_vector_type(8))) float v8f;
typedef unsigned int v4u __attribute__((ext_vector_type(4)));
typedef int v4i __attribute__((ext_vector_type(4)));
typedef int v8i __attribute__((ext_vector_type(8)));

#if __has_builtin(__builtin_amdgcn_tensor_load_to_lds)
#define HAVE_TDM 1
#else
#define HAVE_TDM 0
#endif

// --------------------------- init: pack spins, zero imag/logd --------------
__global__ void k_init(const float* __restrict__ wa, const float* __restrict__ wb,
                       float* __restrict__ Wr, float* __restrict__ Wi,
                       float* __restrict__ logd) {
  int idx = blockIdx.x * blockDim.x + threadIdx.x;   // 0..32767
  int row = idx >> 7, col = idx & 127;
  float v = (col < NE) ? wa[row * NE + col] : wb[row * NE + (col - NE)];
  Wr[idx] = v;
  Wi[idx] = 0.f;
  if (idx < 2) logd[idx] = 0.f;
}

// ------------- build V[t] = sqrt(|ts_v[t]|) * sum_k fields[t,k]*vhs[k] -----
// Bandwidth-bound: one pass over vhs (134 MB), 20 accumulators per element.
__global__ __launch_bounds__(256) void k_build_v(const float* __restrict__ fields,
                                                 const float* __restrict__ vhs,
                                                 const float* __restrict__ ts_v,
                                                 float* __restrict__ V) {
  __shared__ float fs[NT * NF];                      // 40 KB
  int tid = threadIdx.x;
  for (int i = tid; i < NT * NF; i += 256) fs[i] = fields[i];
  __syncthreads();

  int pq = blockIdx.x * 256 + tid;                   // 0..65535
  float acc[NT];
#pragma unroll
  for (int t = 0; t < NT; ++t) acc[t] = 0.f;

  for (int k = 0; k < NF; ++k) {
    float v = vhs[(size_t)k * (NB * NB) + pq];
#pragma unroll
    for (int t = 0; t < NT; ++t) acc[t] = fmaf(fs[t * NF + k], v, acc[t]);
  }
#pragma unroll
  for (int t = 0; t < NT; ++t) {
    float s = sqrtf(fabsf(ts_v[t]));
    V[(size_t)t * (NB * NB) + pq] = s * acc[t];
  }
}

// --------------------------- expm_apply via WMMA f32 -----------------------
// mode 0: operator = c*A (real), c = -ts_h[step]     (scale_p points at ts_h)
// mode 1: operator = i*A (A real)  => (Tr,Ti) -> (-A*Ti, A*Tr) each order
// One block per 16-column slab of W. The full 256x256 operator matrix is
// staged into LDS (pitch 257) once via the Tensor Data Mover; the 256x16
// complex term slab also lives in LDS. Taylor output accumulates in v8f regs.
// Waves 0..15 produce one plane, 16..31 the other.
__global__ __launch_bounds__(1024) void k_expm(const float* __restrict__ A,
                                               float* __restrict__ Wr,
                                               float* __restrict__ Wi,
                                               const float* scale_p, int mode) {
  extern __shared__ float sm[];
  float* As     = sm;                    // NB*LDA floats (257 KB)
  float* term_r = sm + NB * LDA;         // NB*16
  float* term_i = term_r + NB * 16;      // NB*16

  const int tid   = threadIdx.x;
  const int lane  = tid & 31;
  const int wave  = tid >> 5;            // 0..31
  const int half  = wave >> 4;           // 0 / 1
  const int mtile = wave & 15;           // M tile 0..15
  const int c0    = blockIdx.x * 16;     // column slab base
  const int rloc  = lane & 15;           // A-row / B,C,D-col within tile
  const int koff  = (lane >> 4) * 2;     // K pair select (ISA 7.12.2 A layout)
  const int moff  = (lane >> 4) * 8;     // C/D row split at lane 16

#if HAVE_TDM
  // ---- stage A (256x256 f32) into LDS with one TDM op, padded to pitch 257
  if (wave == 0) {
    unsigned long long ga = (unsigned long long)(uintptr_t)A;
    unsigned int lds_off  = (unsigned int)(uintptr_t)As;   // LDS byte offset
    v4u g0;
    g0.x = 1u;                                        // count=1 (valid user D#)
    g0.y = lds_off;                                   // lds_addr
    g0.z = (unsigned int)ga;                          // global_addr[31:0]
    g0.w = (unsigned int)((ga >> 32) & 0x01FFFFFFu)   // global_addr[56:32]
         | 0x80000000u;                               // type=2 ("image")
    v8i g1;
    g1[0] = (int)(0x00020000u                         // data_size = 4 bytes
                | (1u << 20)                          // pad_enable
                | (7u << 22));                        // pad_interval=256 DWORDs
                                                      // pad_amount bits=0 -> 1 DWORD
    g1[1] = (int)(256u << 16);                        // tensor_dim0[15:0]=256
    g1[2] = (int)(256u << 16);                        // tensor_dim1[15:0]=256
    g1[3] = (int)(256u << 16);                        // tile_dim0=256
    g1[4] = 256;                                      // tile_dim1=256, tile_dim2=0
    g1[5] = 256;                                      // tensor_dim0_stride=256
    g1[6] = 0;
    g1[7] = 0;
    v4i gz = {0, 0, 0, 0};
#if defined(__clang_major__) && (__clang_major__ >= 23)
    v8i z8 = {0, 0, 0, 0, 0, 0, 0, 0};
    __builtin_amdgcn_tensor_load_to_lds(g0, g1, gz, gz, z8, 0);
#else
    __builtin_amdgcn_tensor_load_to_lds(g0, g1, gz, gz, 0);
#endif
    __builtin_amdgcn_s_wait_tensorcnt(0);
  }
#else
  for (int i = tid; i < NB * NB; i += 1024)
    As[(i >> 8) * LDA + (i & 255)] = A[i];
#endif

  // ---- load W slab into LDS term planes
  for (int i = tid; i < NB * 16; i += 1024) {
    int row = i >> 4, col = i & 15;
    term_r[i] = Wr[row * NC + c0 + col];
    term_i[i] = Wi[row * NC + c0 + col];
  }
  __syncthreads();

  // plane roles
  const float* srcp;
  float* dstp;
  float sgn;
  float c;
  if (mode == 0) {
    c    = -scale_p[0];
    srcp = half ? term_i : term_r;
    dstp = half ? term_i : term_r;
    sgn  = 1.f;
  } else {
    c = 1.f;                                          // sqrt(ts_v) folded into V
    if (half == 0) { srcp = term_r; dstp = term_i; sgn =  1.f; }
    else           { srcp = term_i; dstp = term_r; sgn = -1.f; }
  }
  const float base = sgn * c;

  // out accumulator (k=0 Taylor term) in registers, positions per C/D layout
  v8f out;
#pragma unroll
  for (int v = 0; v < 8; ++v)
    out[v] = dstp[(mtile * 16 + v + moff) * 16 + rloc];

  const float* ArowL = As + (mtile * 16 + rloc) * LDA + koff;

  // Taylor orders fully unrolled: 1/k folds to an immediate per copy
  // (no SALU branch ladder, no division between barriers)
#pragma unroll
  for (int k = 1; k <= EXPM_ORDER; ++k) {
    v8f acc = {};
#pragma unroll 4
    for (int kk = 0; kk < 64; ++kk) {                 // K = 256 in steps of 4
      v2f a;                                          // A 16x4 frag (ISA 7.12.2)
      a.x = ArowL[kk * 4 + 0];
      a.y = ArowL[kk * 4 + 1];
      v2f b;                                          // B 4x16 frag
      int kb = kk * 4 + koff;
      b.x = srcp[(kb + 0) * 16 + rloc];
      b.y = srcp[(kb + 1) * 16 + rloc];
      acc = __builtin_amdgcn_wmma_f32_16x16x4_f32(
          false, a, false, b, (short)0, acc, false, false);
    }
    __syncthreads();                                  // all src reads done
    float f = base * (1.0f / (float)k);               // compile-time constant
#pragma unroll
    for (int v = 0; v < 8; ++v) {
      float t = acc[v] * f;
      dstp[(mtile * 16 + v + moff) * 16 + rloc] = t;  // new Taylor term
      out[v] += t;
    }
    __syncthreads();                                  // term visible to all
  }

  float* G = (dstp == term_r) ? Wr : Wi;              // global plane we own
#pragma unroll
  for (int v = 0; v < 8; ++v)
    G[(mtile * 16 + v + moff) * NC + c0 + rloc] = out[v];
}

// ------------- per-spin Classical Gram-Schmidt (== phase-fixed QR) ----------
// One block per spin; 256x64 complex panel in dynamic LDS (padded pitch).
__global__ __launch_bounds__(256) void k_ortho(float* __restrict__ Wr,
                                               float* __restrict__ Wi,
                                               float* __restrict__ logd) {
  extern __shared__ float sm[];
  float* qr_  = sm;                  // NB*LDP
  float* qi_  = qr_ + NB * LDP;      // NB*LDP
  float* rre  = qi_ + NB * LDP;      // 64
  float* rim  = rre + 64;            // 64
  float* red  = rim + 64;            // 8
  float* sinv = red + 8;             // 1

  const int s = blockIdx.x;          // spin
  const int tid = threadIdx.x, lane = tid & 31, wave = tid >> 5;

  for (int i = tid; i < NB * NE; i += 256) {
    int row = i >> 6, col = i & 63;
    qr_[row * LDP + col] = Wr[row * NC + s * NE + col];
    qi_[row * LDP + col] = Wi[row * NC + s * NE + col];
  }
  __syncthreads();

  float logsum = 0.f;                // meaningful on tid 0 only
  for (int j = 0; j < NE; ++j) {
    // r_ij = q_i^dagger w_j, i < j  (waves split the i's, lanes split rows)
    for (int i = wave; i < j; i += 8) {
      float sre = 0.f, sim = 0.f;
      for (int r = lane; r < NB; r += 32) {
        float ar = qr_[r * LDP + i], ai = qi_[r * LDP + i];
        float br = qr_[r * LDP + j], bi = qi_[r * LDP + j];
        sre += ar * br + ai * bi;
        sim += ar * bi - ai * br;
      }
#pragma unroll
      for (int off = 16; off > 0; off >>= 1) {
        sre += __shfl_xor(sre, off, 32);
        sim += __shfl_xor(sim, off, 32);
      }
      if (lane == 0) { rre[i] = sre; rim[i] = sim; }
    }
    __syncthreads();

    // w_j -= sum_i r_ij * q_i   (one row per thread)
    int row = tid;
    float wrj = qr_[row * LDP + j], wij = qi_[row * LDP + j];
    for (int i = 0; i < j; ++i) {
      float ar = qr_[row * LDP + i], ai = qi_[row * LDP + i];
      float rr = rre[i], ri = rim[i];
      wrj -= rr * ar - ri * ai;
      wij -= rr * ai + ri * ar;
    }
    qr_[row * LDP + j] = wrj;
    qi_[row * LDP + j] = wij;

    // r_jj = ||w_j||, normalize (CGS => real positive diag, matches sign fix)
    float sq = wrj * wrj + wij * wij;
#pragma unroll
    for (int off = 16; off > 0; off >>= 1) sq += __shfl_xor(sq, off, 32);
    if (lane == 0) red[wave] = sq;
    __syncthreads();
    if (tid == 0) {
      float n2 = 0.f;
      for (int w = 0; w < 8; ++w) n2 += red[w];
      float rjj = sqrtf(n2);
      logsum += logf(rjj);
      sinv[0] = 1.0f / rjj;
    }
    __syncthreads();
    float inv = sinv[0];
    qr_[row * LDP + j] = wrj * inv;
    qi_[row * LDP + j] = wij * inv;
    __syncthreads();
  }

  if (tid == 0) logd[s] += logsum;   // one writer per slot: deterministic
  for (int i = tid; i < NB * NE; i += 256) {
    int row = i >> 6, col = i & 63;
    Wr[row * NC + s * NE + col] = qr_[row * LDP + col];
    Wi[row * NC + s * NE + col] = qi_[row * LDP + col];
  }
}

// --------------------------- pack outputs ----------------------------------
__global__ void k_pack(const float* __restrict__ Wr, const float* __restrict__ Wi,
                       const float* __restrict__ logd, const float* __restrict__ enuc,
                       float* __restrict__ out) {
  int idx = blockIdx.x * 256 + threadIdx.x;          // 0..16383 (row,col) pairs
  int row = idx >> 6, col = idx & 63;
  // wfn_a interleaved complex64
  out[idx * 2 + 0] = Wr[row * NC + col];
  out[idx * 2 + 1] = Wi[row * NC + col];
  // wfn_b
  out[2 * NB * NE + idx * 2 + 0] = Wr[row * NC + NE + col];
  out[2 * NB * NE + idx * 2 + 1] = Wi[row * NC + NE + col];
  if (idx == 0) out[4 * NB * NE] = enuc[0] + logd[0] + logd[1];
}

// ---------------------------------------------------------------------------
extern "C" void kernel_launch(void* const* d_in, const int* in_sizes, int n_in,
                              void* d_out, int out_size, void* d_ws, size_t ws_size,
                              hipStream_t stream) {
  (void)in_sizes; (void)n_in; (void)out_size; (void)ws_size;
  const float* wfn_a  = (const float*)d_in[0];   // (256,64) f32
  const float* wfn_b  = (const float*)d_in[1];   // (256,64) f32
  const float* fields = (const float*)d_in[2];   // (20,512) f32
  const float* hmf    = (const float*)d_in[3];   // (256,256) f32
  const float* vhs    = (const float*)d_in[4];   // (512,256,256) f32
  const float* ts_v   = (const float*)d_in[5];   // (20,) f32
  const float* ts_h   = (const float*)d_in[6];   // (21,) f32
  const float* enuc   = (const float*)d_in[7];   // (1,) f32
  float* out = (float*)d_out;

  // workspace layout (floats)
  float* Wr   = (float*)d_ws;                    // 256*128
  float* Wi   = Wr + NB * NC;                    // 256*128
  float* logd = Wi + NB * NC;                    // 2 (+pad)
  float* V    = logd + 16;                       // 20*256*256  (~5.2 MB)

  const size_t smem_expm  = (size_t)(NB * LDA + 2 * NB * 16) * sizeof(float); // ~289 KB
  const size_t smem_ortho = (size_t)(2 * NB * LDP + 192) * sizeof(float);     // ~130 KB

  k_init<<<(NB * NC) / 256, 256, 0, stream>>>(wfn_a, wfn_b, Wr, Wi, logd);
  k_build_v<<<(NB * NB) / 256, 256, 0, stream>>>(fields, vhs, ts_v, V);

  for (int ii = 0; ii < NT; ++ii) {
    k_expm<<<NC / 16, 1024, smem_expm, stream>>>(hmf, Wr, Wi, ts_h + ii, 0);
    k_expm<<<NC / 16, 1024, smem_expm, stream>>>(V + (size_t)ii * NB * NB, Wr, Wi,
                                                 nullptr, 1);
    if ((ii + 1) % 5 == 0)
      k_ortho<<<2, 256, smem_ortho, stream>>>(Wr, Wi, logd);
  }
  k_expm<<<NC / 16, 1024, smem_expm, stream>>>(hmf, Wr, Wi, ts_h + NT, 0);
  k_pack<<<(NB * NE) / 256, 256, 0, stream>>>(Wr, Wi, logd, enuc, out);
}